// GCNLayer_57071525429600
// MI455X (gfx1250) — compile-verified
//
#include <hip/hip_runtime.h>
#include <cstdint>
#include <cstddef>

// CDNA5 / gfx1250 GCNConv + ReLU
//   out = relu( D^-1/2 (A + I) D^-1/2 (x @ W) + b )
// Phases: degree (int atomics) -> dinv -> fp32 WMMA GEMM with src-side
// normalization folded in -> self-loop init -> edge scatter (native f32
// atomics via inline asm, L2-resident working set) -> bias+relu.

typedef float v2f __attribute__((ext_vector_type(2)));
typedef float v8f __attribute__((ext_vector_type(8)));

#define FEAT 256

// Guaranteed-native non-returning f32 atomic add (STOREcnt tracked; implicit
// wait-idle at s_endpgm covers completion; 'memory' clobber keeps ordering).
__device__ __forceinline__ void atomic_add_f32_native(float* p, float v) {
  asm volatile("global_atomic_add_f32 %0, %1, off"
               :
               : "v"((unsigned long long)(uintptr_t)p), "v"(v)
               : "memory");
}

// ---- degree = 1 (self loop) + incoming edges -------------------------------
__global__ void k_init_deg(int* __restrict__ deg, int n) {
  int i = blockIdx.x * blockDim.x + threadIdx.x;
  if (i < n) deg[i] = 1;
}

__global__ void k_deg_accum(const int* __restrict__ dst, int* __restrict__ deg, int E) {
  int e = blockIdx.x * blockDim.x + threadIdx.x;
  if (e < E) atomicAdd(&deg[dst[e]], 1);
}

__global__ void k_dinv(const int* __restrict__ deg, float* __restrict__ dinv, int n) {
  int i = blockIdx.x * blockDim.x + threadIdx.x;
  if (i < n) dinv[i] = __frsqrt_rn((float)deg[i]);
}

// ---- xw_scaled = (x @ W) * dinv[row]  (fp32 WMMA 16x16x4) ------------------
// Block = 128 threads (4 waves). Block b owns row tile [16b, 16b+16).
// Wave w owns 4 column tiles: cols [64w, 64w+64). N == 50000 = 3125 * 16.
__global__ void __launch_bounds__(128) k_gemm_scale(
    const float* __restrict__ x, const float* __restrict__ W,
    const float* __restrict__ dinv, float* __restrict__ xws) {
  const int lane = threadIdx.x & 31;
  const int wave = threadIdx.x >> 5;
  const int row0 = blockIdx.x * 16;
  const int col0 = wave * 64;
  const int h = lane >> 4;   // lane half selects K pair (A/B) and row half (C/D)
  const int n = lane & 15;   // row index for A, col index for B/C/D

  v8f acc0 = {}, acc1 = {}, acc2 = {}, acc3 = {};

  // A fragment source: row (row0+n), K pair starting at k + 2h
  const float* arow = x + (size_t)(row0 + n) * FEAT + 2 * h;

  for (int k = 0; k < FEAT; k += 4) {
    v2f a = *(const v2f*)(arow + k);  // VGPR0 = K=k+2h, VGPR1 = K=k+2h+1
    const float* wr0 = W + (size_t)(k + 2 * h) * FEAT + col0 + n;
    const float* wr1 = wr0 + FEAT;
    v2f b0; b0.x = wr0[0];  b0.y = wr1[0];
    v2f b1; b1.x = wr0[16]; b1.y = wr1[16];
    v2f b2; b2.x = wr0[32]; b2.y = wr1[32];
    v2f b3; b3.x = wr0[48]; b3.y = wr1[48];
    acc0 = __builtin_amdgcn_wmma_f32_16x16x4_f32(false, a, false, b0, (short)0, acc0, false, false);
    acc1 = __builtin_amdgcn_wmma_f32_16x16x4_f32(false, a, false, b1, (short)0, acc1, false, false);
    acc2 = __builtin_amdgcn_wmma_f32_16x16x4_f32(false, a, false, b2, (short)0, acc2, false, false);
    acc3 = __builtin_amdgcn_wmma_f32_16x16x4_f32(false, a, false, b3, (short)0, acc3, false, false);
  }

  // Epilogue: C/D VGPR r holds rows (r + 8h), col n. Scale by dinv and store.
#pragma unroll
  for (int r = 0; r < 8; ++r) {
    int row = row0 + r + 8 * h;
    float s = dinv[row];
    float* orow = xws + (size_t)row * FEAT + col0 + n;
    orow[0]  = acc0[r] * s;
    orow[16] = acc1[r] * s;
    orow[32] = acc2[r] * s;
    orow[48] = acc3[r] * s;
  }
}

// ---- out = xw_scaled * dinv[row]   (self-loop term, also clears poison) ----
__global__ void k_selfloop(const float* __restrict__ xws, const float* __restrict__ dinv,
                           float* __restrict__ out, int n4) {
  int t = blockIdx.x * blockDim.x + threadIdx.x;  // n4 = N * 64 float4s
  if (t >= n4) return;
  int row = t >> 6;
  float s = dinv[row];
  float4 v = ((const float4*)xws)[t];
  v.x *= s; v.y *= s; v.z *= s; v.w *= s;
  ((float4*)out)[t] = v;
}

// ---- edge scatter: out[dst] += xw_scaled[src] * dinv[dst] ------------------
// One wave per edge: lane l owns 8 contiguous channels [8l, 8l+8).
// Gather = 2x global_load_b128 per lane (wave covers one contiguous 1KB row);
// scatter = 8 native global_atomic_add_f32 per lane.
__global__ void __launch_bounds__(256) k_scatter(
    const int* __restrict__ src, const int* __restrict__ dst,
    const float* __restrict__ xws, const float* __restrict__ dinv,
    float* __restrict__ out, int E) {
  int e = blockIdx.x * 8 + (threadIdx.x >> 5);
  if (e >= E) return;
  int lane = threadIdx.x & 31;
  int s = src[e];
  int d = dst[e];
  float coef = dinv[d];
  const float4* xr = (const float4*)(xws + (size_t)s * FEAT + lane * 8);
  float4 a0 = xr[0];
  float4 a1 = xr[1];
  float* orow = out + (size_t)d * FEAT + lane * 8;
  atomic_add_f32_native(orow + 0, a0.x * coef);
  atomic_add_f32_native(orow + 1, a0.y * coef);
  atomic_add_f32_native(orow + 2, a0.z * coef);
  atomic_add_f32_native(orow + 3, a0.w * coef);
  atomic_add_f32_native(orow + 4, a1.x * coef);
  atomic_add_f32_native(orow + 5, a1.y * coef);
  atomic_add_f32_native(orow + 6, a1.z * coef);
  atomic_add_f32_native(orow + 7, a1.w * coef);
}

// ---- out = relu(out + b) ---------------------------------------------------
__global__ void k_bias_relu(float* __restrict__ out, const float* __restrict__ bias, int n4) {
  int t = blockIdx.x * blockDim.x + threadIdx.x;  // n4 = N * 64 float4s
  if (t >= n4) return;
  float4 v = ((float4*)out)[t];
  float4 b4 = ((const float4*)bias)[t & 63];
  v.x = fmaxf(v.x + b4.x, 0.0f);
  v.y = fmaxf(v.y + b4.y, 0.0f);
  v.z = fmaxf(v.z + b4.z, 0.0f);
  v.w = fmaxf(v.w + b4.w, 0.0f);
  ((float4*)out)[t] = v;
}

extern "C" void kernel_launch(void* const* d_in, const int* in_sizes, int n_in,
                              void* d_out, int out_size, void* d_ws, size_t ws_size,
                              hipStream_t stream) {
  const float* x  = (const float*)d_in[0];
  const int*   ei = (const int*)d_in[1];   // [2, E] row-major: src row then dst row
  const float* W  = (const float*)d_in[2];
  const float* b  = (const float*)d_in[3];
  float* out = (float*)d_out;

  const int N = in_sizes[0] / FEAT;   // 50000
  const int E = in_sizes[1] / 2;      // 1600000
  const int* src = ei;
  const int* dst = ei + E;

  // Workspace layout: deg @ 0 (N ints), dinv @ 1MB (N floats),
  // xw_scaled @ 2MB (N*256 floats ~ 51.2MB). Needs ws_size >= ~54MB.
  char* ws = (char*)d_ws;
  int*   deg  = (int*)ws;
  float* dinv = (float*)(ws + (size_t)(1u << 20));
  float* xws  = (float*)(ws + (size_t)(2u << 20));

  const int T = 256;
  k_init_deg<<<(N + T - 1) / T, T, 0, stream>>>(deg, N);
  k_deg_accum<<<(E + T - 1) / T, T, 0, stream>>>(dst, deg, E);
  k_dinv<<<(N + T - 1) / T, T, 0, stream>>>(deg, dinv, N);

  k_gemm_scale<<<N / 16, 128, 0, stream>>>(x, W, dinv, xws);

  const int n4 = N * (FEAT / 4);
  k_selfloop<<<(n4 + T - 1) / T, T, 0, stream>>>(xws, dinv, out, n4);
  k_scatter<<<(E + 7) / 8, T, 0, stream>>>(src, dst, xws, dinv, out, E);
  k_bias_relu<<<(n4 + T - 1) / T, T, 0, stream>>>(out, b, n4);
}